// global_cluster_pool_85306640433592
// MI455X (gfx1250) — compile-verified
//
#include <hip/hip_runtime.h>
#include <hip/hip_bf16.h>

#define ROI   400
#define HID   256
#define CDIM  64
#define NC    14
#define NB    256
#define NPTS  102400
#define BN_EPS 1e-5f
#define SLOPE  0.01f

typedef __attribute__((ext_vector_type(2))) float v2f;
typedef __attribute__((ext_vector_type(8))) float v8f;

// ---------------------------------------------------------------------------
// K1: cluster assignment. wave32 per row; Wc staged in LDS (14*400*4 = 22.4KB).
// argmax(softmax(logits)) == argmax(logits).
// ---------------------------------------------------------------------------
__global__ void k_assign(const float* __restrict__ pos,
                         const float* __restrict__ Wc,
                         int* __restrict__ ci,
                         float* __restrict__ ci_out)
{
    __shared__ float wc[NC * ROI];
    for (int i = threadIdx.x; i < NC * ROI; i += blockDim.x) wc[i] = Wc[i];
    __syncthreads();

    const int lane = threadIdx.x & 31;
    const int wave = threadIdx.x >> 5;
    const int gw   = blockIdx.x * (blockDim.x >> 5) + wave;
    const int nw   = gridDim.x * (blockDim.x >> 5);

    for (int row = gw; row < NPTS; row += nw) {
        float acc[NC];
#pragma unroll
        for (int c = 0; c < NC; ++c) acc[c] = 0.f;
        const float* p = pos + (size_t)row * ROI;
        for (int k = lane; k < ROI; k += 32) {
            const float pv = p[k];
#pragma unroll
            for (int c = 0; c < NC; ++c) acc[c] += pv * wc[c * ROI + k];
        }
#pragma unroll
        for (int c = 0; c < NC; ++c) {
#pragma unroll
            for (int off = 16; off > 0; off >>= 1)
                acc[c] += __shfl_xor(acc[c], off, 32);
        }
        if (lane == 0) {
            int best = 0; float bv = acc[0];
#pragma unroll
            for (int c = 1; c < NC; ++c)
                if (acc[c] > bv) { bv = acc[c]; best = c; }
            ci[row]     = best;
            ci_out[row] = (float)best;   // second tuple output
        }
    }
}

// ---------------------------------------------------------------------------
// K2: segment-sum into [B, C, H]. batch[] is sorted, so SPLIT blocks per graph
// binary-search the row range, accumulate 14*256 bins in LDS, flush with a few
// global float atomics. Also emits per-graph node counts.
// ---------------------------------------------------------------------------
#define SPLIT 8

__global__ void k_pool(const float* __restrict__ x,
                       const int* __restrict__ batch,
                       const int* __restrict__ ci,
                       float* __restrict__ sums,
                       int* __restrict__ counts)
{
    __shared__ float acc[NC * HID];   // 14 KB
    const int b = blockIdx.x / SPLIT;
    const int s = blockIdx.x % SPLIT;

    int lo = 0, hi = NPTS;
    while (lo < hi) { int m = (lo + hi) >> 1; if (batch[m] < b) lo = m + 1; else hi = m; }
    const int start = lo;
    hi = NPTS;
    while (lo < hi) { int m = (lo + hi) >> 1; if (batch[m] < b + 1) lo = m + 1; else hi = m; }
    const int end = lo;

    if (s == 0 && threadIdx.x == 0) counts[b] = end - start;

    for (int i = threadIdx.x; i < NC * HID; i += blockDim.x) acc[i] = 0.f;
    __syncthreads();

    const int t = threadIdx.x;                 // owns column t exclusively
    for (int r = start + s; r < end; r += SPLIT) {
        const int c = ci[r];                   // wave-uniform load
        acc[c * HID + t] += x[(size_t)r * HID + t];
    }
    __syncthreads();

    float* dst = sums + (size_t)b * NC * HID;
    for (int i = t; i < NC * HID; i += blockDim.x)
        atomicAdd(&dst[i], acc[i]);
}

// ---------------------------------------------------------------------------
// K3: per-cluster GEMM1 with V_WMMA_F32_16X16X4_F32 (full fp32 matrix pipe).
// One wave32 per 16x16 output tile, K-loop of 64 WMMAs. 1/count folded into A,
// bias folded into epilogue. h1[b][c][o].
// ---------------------------------------------------------------------------
__global__ void k_gemm1(const float* __restrict__ sums,
                        const int* __restrict__ counts,
                        const float* __restrict__ W1,
                        const float* __restrict__ b1,
                        float* __restrict__ h1)
{
    const int lane = threadIdx.x & 31;
    const int wave = threadIdx.x >> 5;
    const int tile = blockIdx.x * (blockDim.x >> 5) + wave;  // 14*16*16 tiles
    const int c  = tile >> 8;
    const int r  = tile & 255;
    const int b0 = (r >> 4) << 4;
    const int o0 = (r & 15) << 4;

    const int m  = lane & 15;
    const int hh = lane >> 4;                 // 0 or 1
    const int brow = b0 + m;
    const float invc = 1.0f / fmaxf((float)counts[brow], 1.0f);

    const float* arow = sums + ((size_t)brow * NC + c) * HID;
    const float* bcol = W1 + ((size_t)c * HID + (o0 + m)) * HID;

    v8f acc = {};
    for (int k0 = 0; k0 < HID; k0 += 4) {
        const int ka = k0 + hh * 2;
        v2f a = *(const v2f*)(arow + ka);
        a.x *= invc; a.y *= invc;
        const v2f bb = *(const v2f*)(bcol + ka);
        acc = __builtin_amdgcn_wmma_f32_16x16x4_f32(
                  false, a, false, bb, (short)0, acc, false, false);
    }

    const int o = o0 + m;
    const float bias = b1[c * HID + o];
#pragma unroll
    for (int v = 0; v < 8; ++v) {
        const int br = b0 + v + hh * 8;
        h1[((size_t)br * NC + c) * HID + o] = acc[v] + bias;
    }
}

// ---------------------------------------------------------------------------
// K4: BatchNorm1d (training stats over batch axis, biased var) + LeakyReLU,
// in place on h1. One wave32 per (c,o): 8 elements/lane, shuffle reduce.
// ---------------------------------------------------------------------------
__global__ void k_bn1(float* __restrict__ h1,
                      const float* __restrict__ g1,
                      const float* __restrict__ be1)
{
    const int lane = threadIdx.x & 31;
    const int wave = threadIdx.x >> 5;
    const int pair = blockIdx.x * (blockDim.x >> 5) + wave;  // 14*256 pairs
    const int c = pair >> 8;
    const int o = pair & 255;

    const size_t stride = (size_t)NC * HID;
    const size_t base = (size_t)c * HID + o;

    float s = 0.f, ss = 0.f, vals[8];
#pragma unroll
    for (int i = 0; i < 8; ++i) {
        const float v = h1[base + (size_t)(lane + i * 32) * stride];
        vals[i] = v; s += v; ss += v * v;
    }
#pragma unroll
    for (int off = 16; off > 0; off >>= 1) {
        s  += __shfl_xor(s, off, 32);
        ss += __shfl_xor(ss, off, 32);
    }
    const float mu  = s * (1.f / NB);
    const float var = ss * (1.f / NB) - mu * mu;
    const float k   = g1[base] * rsqrtf(var + BN_EPS);
    const float bet = be1[base];
#pragma unroll
    for (int i = 0; i < 8; ++i) {
        float hv = (vals[i] - mu) * k + bet;
        hv = hv >= 0.f ? hv : SLOPE * hv;
        h1[base + (size_t)(lane + i * 32) * stride] = hv;
    }
}

// ---------------------------------------------------------------------------
// K5: per-cluster GEMM2 (HID -> CDIM) with fp32 WMMA. h2[b][c][o2].
// ---------------------------------------------------------------------------
__global__ void k_gemm2(const float* __restrict__ h1,
                        const float* __restrict__ W2,
                        const float* __restrict__ b2,
                        float* __restrict__ h2)
{
    const int lane = threadIdx.x & 31;
    const int wave = threadIdx.x >> 5;
    const int tile = blockIdx.x * (blockDim.x >> 5) + wave;  // 14*16*4 tiles
    const int c  = tile >> 6;
    const int r  = tile & 63;
    const int b0 = (r >> 2) << 4;
    const int o0 = (r & 3) << 4;

    const int m  = lane & 15;
    const int hh = lane >> 4;
    const int brow = b0 + m;

    const float* arow = h1 + ((size_t)brow * NC + c) * HID;
    const float* bcol = W2 + ((size_t)c * CDIM + (o0 + m)) * HID;

    v8f acc = {};
    for (int k0 = 0; k0 < HID; k0 += 4) {
        const int ka = k0 + hh * 2;
        const v2f a  = *(const v2f*)(arow + ka);
        const v2f bb = *(const v2f*)(bcol + ka);
        acc = __builtin_amdgcn_wmma_f32_16x16x4_f32(
                  false, a, false, bb, (short)0, acc, false, false);
    }

    const int o = o0 + m;
    const float bias = b2[c * CDIM + o];
#pragma unroll
    for (int v = 0; v < 8; ++v) {
        const int br = b0 + v + hh * 8;
        h2[((size_t)br * NC + c) * CDIM + o] = acc[v] + bias;
    }
}

// ---------------------------------------------------------------------------
// K6: BN2 + LeakyReLU, writing the concatenated output [B, C*CDIM].
// ---------------------------------------------------------------------------
__global__ void k_bn2(const float* __restrict__ h2,
                      const float* __restrict__ g2,
                      const float* __restrict__ be2,
                      float* __restrict__ out)
{
    const int lane = threadIdx.x & 31;
    const int wave = threadIdx.x >> 5;
    const int pair = blockIdx.x * (blockDim.x >> 5) + wave;  // 14*64 pairs
    const int c = pair >> 6;
    const int o = pair & 63;

    const size_t stride = (size_t)NC * CDIM;   // 896
    const size_t base = (size_t)c * CDIM + o;

    float s = 0.f, ss = 0.f, vals[8];
#pragma unroll
    for (int i = 0; i < 8; ++i) {
        const float v = h2[base + (size_t)(lane + i * 32) * stride];
        vals[i] = v; s += v; ss += v * v;
    }
#pragma unroll
    for (int off = 16; off > 0; off >>= 1) {
        s  += __shfl_xor(s, off, 32);
        ss += __shfl_xor(ss, off, 32);
    }
    const float mu  = s * (1.f / NB);
    const float var = ss * (1.f / NB) - mu * mu;
    const float k   = g2[base] * rsqrtf(var + BN_EPS);
    const float bet = be2[base];
#pragma unroll
    for (int i = 0; i < 8; ++i) {
        float hv = (vals[i] - mu) * k + bet;
        hv = hv >= 0.f ? hv : SLOPE * hv;
        out[(size_t)(lane + i * 32) * stride + base] = hv;  // out[b, c*64+o]
    }
}

// ---------------------------------------------------------------------------
extern "C" void kernel_launch(void* const* d_in, const int* in_sizes, int n_in,
                              void* d_out, int out_size, void* d_ws, size_t ws_size,
                              hipStream_t stream)
{
    // inputs: x, pos, Wc, W1, b1, g1, be1, W2, b2, g2, be2, batch, size
    const float* x     = (const float*)d_in[0];
    const float* pos   = (const float*)d_in[1];
    const float* Wc    = (const float*)d_in[2];
    const float* W1    = (const float*)d_in[3];
    const float* b1    = (const float*)d_in[4];
    const float* g1    = (const float*)d_in[5];
    const float* be1   = (const float*)d_in[6];
    const float* W2    = (const float*)d_in[7];
    const float* b2    = (const float*)d_in[8];
    const float* g2    = (const float*)d_in[9];
    const float* be2   = (const float*)d_in[10];
    const int*   batch = (const int*)d_in[11];

    float* out  = (float*)d_out;             // [256, 896]
    float* ci_f = out + (size_t)NB * NC * CDIM;  // [N] argmax as float

    // workspace layout
    float* sums   = (float*)d_ws;                         // B*C*H = 917504 f32
    int*   ci     = (int*)(sums + (size_t)NB * NC * HID); // N ints
    int*   counts = ci + NPTS;                            // B ints
    float* h1     = (float*)(counts + NB);                // B*C*H f32
    float* h2     = h1 + (size_t)NB * NC * HID;           // B*C*CDIM f32

    hipMemsetAsync(sums, 0, (size_t)NB * NC * HID * sizeof(float), stream);

    k_assign<<<1024, 256, 0, stream>>>(pos, Wc, ci, ci_f);
    k_pool  <<<NB * SPLIT, 256, 0, stream>>>(x, batch, ci, sums, counts);
    k_gemm1 <<<(NC * 16 * 16) / 8, 256, 0, stream>>>(sums, counts, W1, b1, h1);
    k_bn1   <<<(NC * HID) / 8, 256, 0, stream>>>(h1, g1, be1);
    k_gemm2 <<<(NC * 16 * 4) / 8, 256, 0, stream>>>(h1, W2, b2, h2);
    k_bn2   <<<(NC * CDIM) / 8, 256, 0, stream>>>(h2, g2, be2, out);
}